// Multiplicative_CrossEntropy_18811956756525
// MI455X (gfx1250) — compile-verified
//
#include <hip/hip_runtime.h>

typedef float v4f __attribute__((ext_vector_type(4)));

#define NMODAL 3
#define NCLASS 1024
#define TPB    256
#define NWAVES (TPB / 32)

static __device__ __forceinline__ float block_max(float v, float* scratch, int lane, int wid) {
#pragma unroll
    for (int off = 16; off > 0; off >>= 1)
        v = fmaxf(v, __shfl_xor(v, off, 32));
    if (lane == 0) scratch[wid] = v;
    __syncthreads();
    float r = scratch[0];
#pragma unroll
    for (int i = 1; i < NWAVES; ++i) r = fmaxf(r, scratch[i]);
    return r;
}

static __device__ __forceinline__ float block_sum(float v, float* scratch, int lane, int wid) {
#pragma unroll
    for (int off = 16; off > 0; off >>= 1)
        v += __shfl_xor(v, off, 32);
    if (lane == 0) scratch[wid] = v;
    __syncthreads();
    float r = scratch[0];
#pragma unroll
    for (int i = 1; i < NWAVES; ++i) r += scratch[i];
    return r;
}

// One block per batch row. 256 threads x 4 classes = 1024 classes.
// CDNA5 async-tensor path: stage 3 modality rows (12KB) into LDS with
// global_load_async_to_lds_b128 (non-temporal: data is read exactly once),
// wait on ASYNCcnt, then compute fused stable log-softmax + multiplicative
// weighting entirely from LDS/registers; stream the [B,C] output with NT stores.
__global__ __launch_bounds__(TPB) void mce_kernel(const float* __restrict__ logits,
                                                  const int* __restrict__ target,
                                                  float* __restrict__ out_loss,
                                                  float* __restrict__ out_slp,
                                                  int batch) {
    __shared__ float buf[NMODAL][NCLASS];        // 12 KB staging
    __shared__ float red[2 * NMODAL][NWAVES];    // per-reduction scratch

    const int b    = blockIdx.x;
    const int t    = threadIdx.x;
    const int lane = t & 31;
    const int wid  = t >> 5;

    const unsigned voff     = (unsigned)t * 16u;                       // 16B per lane
    const unsigned lds_base = (unsigned)(uintptr_t)(&buf[0][0]);       // group-relative LDS byte addr

    // ---- async global -> LDS staging (ASYNCcnt path), non-temporal reads ----
#pragma unroll
    for (int m = 0; m < NMODAL; ++m) {
        unsigned long long gbase =
            (unsigned long long)(uintptr_t)(logits + ((size_t)m * (size_t)batch + (size_t)b) * NCLASS);
        unsigned lds_addr = lds_base + (unsigned)m * (NCLASS * 4u) + voff;
        asm volatile("global_load_async_to_lds_b128 %0, %1, %2 th:TH_LOAD_NT"
                     :
                     : "v"(lds_addr), "v"(voff), "s"(gbase)
                     : "memory");
    }
    asm volatile("s_wait_asynccnt 0x0" ::: "memory");

    float acc0 = 0.f, acc1 = 0.f, acc2 = 0.f, acc3 = 0.f;

#pragma unroll
    for (int m = 0; m < NMODAL; ++m) {
        v4f x = *(const v4f*)(&buf[m][t * 4]);

        float lmax = fmaxf(fmaxf(x.x, x.y), fmaxf(x.z, x.w));
        float rmax = block_max(lmax, red[2 * m], lane, wid);

        float e0 = __expf(x.x - rmax);
        float e1 = __expf(x.y - rmax);
        float e2 = __expf(x.z - rmax);
        float e3 = __expf(x.w - rmax);

        float rsum = block_sum(e0 + e1 + e2 + e3, red[2 * m + 1], lane, wid);

        float logZ = rmax + __logf(rsum);
        float inv  = 1.0f / rsum;

        float lp0 = x.x - logZ, lp1 = x.y - logZ, lp2 = x.z - logZ, lp3 = x.w - logZ;
        // exponent = BETA/(M-1) = 2/2 = 1  =>  weighting = softmax = e * inv
        acc0 += (e0 * inv) * lp0;
        acc1 += (e1 * inv) * lp1;
        acc2 += (e2 * inv) * lp2;
        acc3 += (e3 * inv) * lp3;
    }

    // Streaming (write-once) output: nontemporal 128-bit store.
    v4f o;
    o.x = acc0; o.y = acc1; o.z = acc2; o.w = acc3;
    __builtin_nontemporal_store(o, (v4f*)(out_slp + (size_t)b * NCLASS) + t);

    // Loss gather: exactly one thread/component matches target[b] in [0, NCLASS).
    const int tgt  = target[b];
    const int base = t * 4;
    if (tgt == base + 0) out_loss[b] = -acc0;
    if (tgt == base + 1) out_loss[b] = -acc1;
    if (tgt == base + 2) out_loss[b] = -acc2;
    if (tgt == base + 3) out_loss[b] = -acc3;
}

extern "C" void kernel_launch(void* const* d_in, const int* in_sizes, int n_in,
                              void* d_out, int out_size, void* d_ws, size_t ws_size,
                              hipStream_t stream) {
    const float* logits = (const float*)d_in[0];   // [3, B, 1024] f32
    const int*   target = (const int*)d_in[1];     // [B] i32
    const int    batch  = in_sizes[1];

    float* out_loss = (float*)d_out;               // [B]
    float* out_slp  = (float*)d_out + batch;       // [B, 1024]

    mce_kernel<<<dim3((unsigned)batch), dim3(TPB), 0, stream>>>(logits, target, out_loss, out_slp, batch);
}